// MacroAverageMAELoss_27513560498652
// MI455X (gfx1250) — compile-verified
//
#include <hip/hip_runtime.h>
#include <hip/hip_bf16.h>

typedef __attribute__((ext_vector_type(2))) float v2f;
typedef __attribute__((ext_vector_type(8))) float v8f;

#define NCLS   6
#define SLOTS  16   // 16 accumulation slots to spread global atomics
#define COLS   16   // cols 0..5 = sums, 6..11 = counts, 12..15 = zero pad
#define PBLOCKS 2048
#define PTHREADS 256

// ---------------------------------------------------------------- zero ws
__global__ void mae_zero_ws(float* __restrict__ ws) {
    ws[threadIdx.x] = 0.0f;   // 1 block x 256 threads == SLOTS*COLS floats
}

// ------------------------------------------------------- per-element accum
__device__ __forceinline__ void mae_acc(float xv, float tv,
                                        float* s, float* c, float& tot) {
    float ad = __builtin_fabsf(xv - tv);
    tot += ad;
#pragma unroll
    for (int k = 0; k < 5; ++k) {            // class 5 recovered by subtraction
        float m = (tv == (float)k) ? 1.0f : 0.0f;
        s[k] = __builtin_fmaf(m, ad, s[k]);
        c[k] += m;
    }
}

// ---------------------------------------------------------------- partial
__global__ void mae_partial(const float* __restrict__ x,
                            const float* __restrict__ t,
                            float* __restrict__ ws, int n) {
    float s[5]  = {0.f, 0.f, 0.f, 0.f, 0.f};
    float cn[5] = {0.f, 0.f, 0.f, 0.f, 0.f};
    float tot = 0.0f;
    int nproc = 0;

    const int tid    = blockIdx.x * blockDim.x + threadIdx.x;
    const int stride = gridDim.x * blockDim.x;
    const int n4     = n >> 2;

    const float4* x4 = (const float4*)x;
    const float4* t4 = (const float4*)t;

    for (int i = tid; i < n4; i += stride) {
        float4 xv = x4[i];
        float4 tv = t4[i];
        mae_acc(xv.x, tv.x, s, cn, tot);
        mae_acc(xv.y, tv.y, s, cn, tot);
        mae_acc(xv.z, tv.z, s, cn, tot);
        mae_acc(xv.w, tv.w, s, cn, tot);
        nproc += 4;
    }
    for (int i = (n4 << 2) + tid; i < n; i += stride) {   // tail (none for 2^25)
        mae_acc(x[i], t[i], s, cn, tot);
        nproc += 1;
    }

    // Recover class 5 (exact: counts are small integers in fp32)
    float part[12];
    float ssum = 0.f, csum = 0.f;
#pragma unroll
    for (int k = 0; k < 5; ++k) {
        part[k]     = s[k];
        part[6 + k] = cn[k];
        ssum += s[k];
        csum += cn[k];
    }
    part[5]  = tot - ssum;
    part[11] = (float)nproc - csum;

    // wave32 reduction
#pragma unroll
    for (int off = 16; off > 0; off >>= 1) {
#pragma unroll
        for (int j = 0; j < 12; ++j)
            part[j] += __shfl_xor(part[j], off);
    }

    __shared__ float lacc[12];
    if (threadIdx.x < 12) lacc[threadIdx.x] = 0.0f;
    __syncthreads();
    if ((threadIdx.x & 31) == 0) {
#pragma unroll
        for (int j = 0; j < 12; ++j)
            atomicAdd(&lacc[j], part[j]);
    }
    __syncthreads();
    if (threadIdx.x < 12) {
        const int slot = blockIdx.x & (SLOTS - 1);
        atomicAdd(ws + slot * COLS + threadIdx.x, lacc[threadIdx.x]);
    }
}

// ------------------------------------------------- finalize via f32 WMMA
// Column sums of the 16x16 partial table using V_WMMA_F32_16X16X4_F32 with
// an all-ones A (exact fp32 adds, layout-invariant in M/K). 1 wave, EXEC all 1s.
__global__ void mae_finalize(const float* __restrict__ ws,
                             float* __restrict__ out) {
    const int lane = threadIdx.x;      // 0..31
    const int n    = lane & 15;        // column N
    const int kh   = lane >> 4;        // K half: 0 -> K{0,1}, 1 -> K{2,3}

    v2f a;  a.x = 1.0f; a.y = 1.0f;    // all-ones 16x4 A matrix
    v8f acc = {};

#pragma unroll
    for (int g = 0; g < 4; ++g) {      // 4 slot-groups of 4 rows -> all 16 slots
        v2f b;
        b.x = ws[(g * 4 + kh * 2 + 0) * COLS + n];
        b.y = ws[(g * 4 + kh * 2 + 1) * COLS + n];
        acc = __builtin_amdgcn_wmma_f32_16x16x4_f32(
            /*neg_a=*/false, a, /*neg_b=*/false, b,
            /*c_mod=*/(short)0, acc, /*reuse_a=*/false, /*reuse_b=*/false);
    }

    // Every row of D equals the column sums; lanes 0..15 hold columns 0..15.
    __shared__ float col[16];
    if (lane < 16) col[lane] = acc[0];
    __syncthreads();
    if (lane == 0) {
        float r = 0.0f;
#pragma unroll
        for (int c = 0; c < NCLS; ++c) {
            float sc = col[c];
            float cc = col[NCLS + c];
            r += (cc > 0.0f) ? (sc / cc) : 0.0f;
        }
        out[0] = r * (1.0f / (float)NCLS);
    }
}

// ---------------------------------------------------------------- launch
extern "C" void kernel_launch(void* const* d_in, const int* in_sizes, int n_in,
                              void* d_out, int out_size, void* d_ws, size_t ws_size,
                              hipStream_t stream) {
    const float* x = (const float*)d_in[0];
    const float* t = (const float*)d_in[1];
    const int n    = in_sizes[0];
    float* ws      = (float*)d_ws;
    float* out     = (float*)d_out;

    mae_zero_ws<<<1, SLOTS * COLS, 0, stream>>>(ws);
    mae_partial<<<PBLOCKS, PTHREADS, 0, stream>>>(x, t, ws, n);
    mae_finalize<<<1, 32, 0, stream>>>(ws, out);
}